// TemporalConsistencyLoss_40716289966150
// MI455X (gfx1250) — compile-verified
//
#include <hip/hip_runtime.h>
#include <math.h>

typedef __attribute__((ext_vector_type(2))) float v2f;
typedef __attribute__((ext_vector_type(8))) float v8f;

#define BATCH 64
#define KSLOT 32
#define DDIM  256
#define NMASK 16384
#define CHUNKS 64

__device__ __forceinline__ float waveReduceSum(float x) {
  #pragma unroll
  for (int off = 16; off > 0; off >>= 1) x += __shfl_xor(x, off, 32);
  return x;
}

// ---------------------------------------------------------------------------
// Kernel 1: row squared-norms of slots_t (rows 0..2047) and slots_t1 (2048..4095)
// one wave per row, float4 (b128) lane-strided loads + shfl reduction
// ---------------------------------------------------------------------------
__global__ void row_norms_kernel(const float* __restrict__ a,
                                 const float* __restrict__ b,
                                 float* __restrict__ normA,
                                 float* __restrict__ normB) {
  int lane = threadIdx.x & 31;
  int wave = threadIdx.x >> 5;
  int row  = blockIdx.x * 8 + wave;               // 0..4095
  const float* src = (row < BATCH * KSLOT) ? (a + (size_t)row * DDIM)
                                           : (b + (size_t)(row - BATCH * KSLOT) * DDIM);
  const float4* src4 = reinterpret_cast<const float4*>(src);
  float s = 0.f;
  #pragma unroll
  for (int i = 0; i < DDIM / 128; ++i) {          // 2 x b128 per lane
    float4 v = src4[lane + i * 32];
    s += v.x * v.x + v.y * v.y + v.z * v.z + v.w * v.w;
  }
  s = waveReduceSum(s);
  if (lane == 0) {
    if (row < BATCH * KSLOT) normA[row] = s;
    else                     normB[row - BATCH * KSLOT] = s;
  }
}

// ---------------------------------------------------------------------------
// Kernel 2: cost[b] = sqrt(clip(aa + bb - 2*A·B^T, 0)) via V_WMMA_F32_16X16X4_F32
// one wave per 16x16 tile; 64 batches x 4 tiles = 256 waves; 64 WMMAs per wave
// b64 loads match the 16x4 f32 fragment layout (lane holds K pair {2*hi, 2*hi+1})
// ---------------------------------------------------------------------------
__global__ void cost_wmma_kernel(const float* __restrict__ a,
                                 const float* __restrict__ bmat,
                                 const float* __restrict__ normA,
                                 const float* __restrict__ normB,
                                 float* __restrict__ cost) {
  int lane = threadIdx.x;              // 0..31, EXEC all ones (block == 1 wave)
  int b    = blockIdx.x >> 2;
  int tile = blockIdx.x & 3;
  int ti = tile >> 1, tj = tile & 1;
  int hi  = lane >> 4;                 // 0: K=0,1   1: K=2,3
  int l15 = lane & 15;

  const v2f* A2 = reinterpret_cast<const v2f*>(a    + (size_t)(b * KSLOT + ti * 16 + l15) * DDIM);
  const v2f* B2 = reinterpret_cast<const v2f*>(bmat + (size_t)(b * KSLOT + tj * 16 + l15) * DDIM);

  v8f acc = {};
  #pragma unroll 8
  for (int k0 = 0; k0 < DDIM; k0 += 4) {
    v2f av = A2[(k0 >> 1) + hi];
    v2f bv = B2[(k0 >> 1) + hi];
    acc = __builtin_amdgcn_wmma_f32_16x16x4_f32(
        /*neg_a=*/false, av, /*neg_b=*/false, bv,
        /*c_mod=*/(short)0, acc, /*reuse_a=*/false, /*reuse_b=*/false);
  }

  // C/D layout: VGPR r -> M = r + 8*hi, N = lane&15
  int n = tj * 16 + l15;
  float bb = normB[b * KSLOT + n];
  #pragma unroll
  for (int r = 0; r < 8; ++r) {
    int m = ti * 16 + r + hi * 8;
    float aa = normA[b * KSLOT + m];
    float c = aa + bb - 2.0f * acc[r];
    cost[(size_t)(b * KSLOT + m) * KSLOT + n] = sqrtf(fmaxf(c, 0.0f));
  }
}

// ---------------------------------------------------------------------------
// Kernel 3: Jonker-Volgenant shortest augmenting path, one wave32 per batch.
// Lane L owns column j = L+1; scalar state (j==0, augmentation) on lane 0.
// ---------------------------------------------------------------------------
__global__ void hungarian_kernel(const float* __restrict__ cost,
                                 int* __restrict__ match) {
  __shared__ float u[KSLOT + 1], v[KSLOT + 1], minv[KSLOT + 1];
  __shared__ int   p[KSLOT + 1], way[KSLOT + 1], used[KSLOT + 1];
  int lane = threadIdx.x;              // 0..31
  int b    = blockIdx.x;
  const float* C = cost + (size_t)b * KSLOT * KSLOT;

  int j = lane + 1;
  u[j] = 0.f; v[j] = 0.f; p[j] = 0; way[j] = 0;
  if (lane == 0) { u[0] = 0.f; v[0] = 0.f; p[0] = 0; way[0] = 0; }
  __syncthreads();

  for (int i = 1; i <= KSLOT; ++i) {
    if (lane == 0) { p[0] = i; used[0] = 0; }
    minv[j] = __builtin_inff();
    used[j] = 0;
    __syncthreads();

    int j0 = 0;
    while (true) {
      if (lane == 0) used[j0] = 1;
      __syncthreads();

      int   i0  = p[j0];
      float ui0 = u[i0];
      int usedj = used[j];
      float cur = C[(i0 - 1) * KSLOT + (j - 1)] - ui0 - v[j];
      if (!usedj && cur < minv[j]) { minv[j] = cur; way[j] = j0; }

      // argmin over unused columns (lowest index wins ties, like np.argmin)
      float m   = usedj ? __builtin_inff() : minv[j];
      int   idx = j;
      #pragma unroll
      for (int off = 16; off > 0; off >>= 1) {
        float m2 = __shfl_xor(m, off, 32);
        int   i2 = __shfl_xor(idx, off, 32);
        if (m2 < m || (m2 == m && i2 < idx)) { m = m2; idx = i2; }
      }
      int   j1    = idx;
      float delta = m;
      __syncthreads();

      // p[j] distinct across used j (partial matching), p[0]=i distinct from
      // all matched rows -> LDS scatter is conflict-free
      if (usedj) { u[p[j]] += delta; v[j] -= delta; }
      else       { minv[j] -= delta; }
      if (lane == 0) { u[p[0]] += delta; v[0] -= delta; }
      __syncthreads();

      j0 = j1;
      if (p[j0] == 0) break;
    }

    if (lane == 0) {
      int jj = j0;
      while (jj != 0) { int jn = way[jj]; p[jj] = p[jn]; jj = jn; }
    }
    __syncthreads();
  }
  match[b * KSLOT + (p[j] - 1)] = j - 1;
}

// ---------------------------------------------------------------------------
// Kernel 4: loss[b,k] = ||slots_t[b,k] - slots_t1[b,match[b,k]]||^2 (b128 loads)
// ---------------------------------------------------------------------------
__global__ void loss_kernel(const float* __restrict__ a,
                            const float* __restrict__ bmat,
                            const int* __restrict__ match,
                            float* __restrict__ lossv) {
  int lane = threadIdx.x & 31;
  int wave = threadIdx.x >> 5;
  int row  = blockIdx.x * 8 + wave;    // b*KSLOT + k, 0..2047
  int m = match[row];
  int b = row >> 5;
  const float4* pa = reinterpret_cast<const float4*>(a    + (size_t)row * DDIM);
  const float4* pb = reinterpret_cast<const float4*>(bmat + (size_t)(b * KSLOT + m) * DDIM);
  float s = 0.f;
  #pragma unroll
  for (int i = 0; i < DDIM / 128; ++i) {
    float4 va = pa[lane + i * 32];
    float4 vb = pb[lane + i * 32];
    float dx = va.x - vb.x, dy = va.y - vb.y, dz = va.z - vb.z, dw = va.w - vb.w;
    s += dx * dx + dy * dy + dz * dz + dw * dw;
  }
  s = waveReduceSum(s);
  if (lane == 0) lossv[row] = s;
}

// ---------------------------------------------------------------------------
// Kernel 5: masks_t.sum(axis=1) stage 1 — the roofline-dominant 134MB stream.
// Per-lane b128 loads: a wave covers 4 contiguous rows (512B) per iteration,
// a block covers 32 contiguous rows (4KB). 4x fewer VMEM instructions than b32.
// Deterministic fixed-order LDS reduction (no float atomics).
// ---------------------------------------------------------------------------
__global__ void mask_partial_kernel(const float* __restrict__ masks,
                                    float* __restrict__ partial) {
  __shared__ float red[256][4];
  int t    = threadIdx.x;
  int lane = t & 31;
  int w    = t >> 5;                   // wave 0..7
  int b    = blockIdx.x / CHUNKS;
  int ch   = blockIdx.x % CHUNKS;
  const int rowsPerChunk = NMASK / CHUNKS;       // 256
  const float4* base = reinterpret_cast<const float4*>(masks + (size_t)b * NMASK * KSLOT);
  int kg = lane & 7;                   // float4 group within a 32-float row
  int ro = lane >> 3;                  // row offset 0..3 within the wave
  float4 acc = make_float4(0.f, 0.f, 0.f, 0.f);
  #pragma unroll 4
  for (int i = 0; i < rowsPerChunk / 32; ++i) {  // 8 iterations
    int n = ch * rowsPerChunk + i * 32 + w * 4 + ro;
    float4 vv = base[(size_t)n * 8 + kg];        // global_load_b128, coalesced
    acc.x += vv.x; acc.y += vv.y; acc.z += vv.z; acc.w += vv.w;
  }
  reinterpret_cast<float4*>(red)[t] = acc;       // ds_store_b128
  __syncthreads();
  if (t < KSLOT) {                               // t == k
    int kg2 = t >> 2;
    int c   = t & 3;
    float s = 0.f;
    #pragma unroll
    for (int ww = 0; ww < 8; ++ww)
      #pragma unroll
      for (int g = 0; g < 4; ++g)
        s += red[ww * 32 + g * 8 + kg2][c];
    partial[(size_t)blockIdx.x * KSLOT + t] = s;
  }
}

// Kernel 6: stage 2 — util[b,k] = sum over chunks (fixed order => deterministic)
__global__ void util_reduce_kernel(const float* __restrict__ partial,
                                   float* __restrict__ util) {
  int idx = blockIdx.x * blockDim.x + threadIdx.x;   // b*KSLOT + k
  if (idx >= BATCH * KSLOT) return;
  int b = idx >> 5, k = idx & 31;
  float s = 0.f;
  for (int c = 0; c < CHUNKS; ++c) s += partial[(size_t)(b * CHUNKS + c) * KSLOT + k];
  util[idx] = s;
}

// Kernel 7: final = mean over (b,k) of loss[b,k] * util[b,k] / sum_k util[b,k]
__global__ void final_kernel(const float* __restrict__ lossv,
                             const float* __restrict__ util,
                             float* __restrict__ out) {
  __shared__ float red[BATCH];
  int b = threadIdx.x;                 // 64 threads
  float s = 0.f, l = 0.f;
  for (int k = 0; k < KSLOT; ++k) {
    float uu = util[b * KSLOT + k];
    s += uu;
    l += lossv[b * KSLOT + k] * uu;
  }
  red[b] = l / s;
  __syncthreads();
  if (b == 0) {
    float tot = 0.f;
    for (int i = 0; i < BATCH; ++i) tot += red[i];
    out[0] = tot / (float)(BATCH * KSLOT);
  }
}

// ---------------------------------------------------------------------------
extern "C" void kernel_launch(void* const* d_in, const int* in_sizes, int n_in,
                              void* d_out, int out_size, void* d_ws, size_t ws_size,
                              hipStream_t stream) {
  const float* slots_t  = (const float*)d_in[0];   // [64,32,256]
  const float* slots_t1 = (const float*)d_in[1];   // [64,32,256]
  const float* masks_t  = (const float*)d_in[2];   // [64,16384,32]
  float* out = (float*)d_out;

  float* ws      = (float*)d_ws;
  float* cost    = ws;                 //  65536 f
  float* normA   = ws + 65536;         //   2048 f
  float* normB   = ws + 67584;         //   2048 f
  int*   match   = (int*)(ws + 69632); //   2048 i
  float* lossv   = ws + 71680;         //   2048 f
  float* partial = ws + 73728;         // 131072 f
  float* util    = ws + 204800;        //   2048 f  (total ~808 KB)

  // dominant memory-bound kernel first
  mask_partial_kernel<<<BATCH * CHUNKS, 256, 0, stream>>>(masks_t, partial);
  util_reduce_kernel<<<8, 256, 0, stream>>>(partial, util);

  row_norms_kernel<<<512, 256, 0, stream>>>(slots_t, slots_t1, normA, normB);
  cost_wmma_kernel<<<BATCH * 4, 32, 0, stream>>>(slots_t, slots_t1, normA, normB, cost);
  hungarian_kernel<<<BATCH, 32, 0, stream>>>(cost, match);
  loss_kernel<<<256, 256, 0, stream>>>(slots_t, slots_t1, match, lossv);
  final_kernel<<<1, BATCH, 0, stream>>>(lossv, util, out);
}